// MultiHeadGATLayer_3951369912901
// MI455X (gfx1250) — compile-verified
//
#include <hip/hip_runtime.h>
#include <hip/hip_bf16.h>

typedef __attribute__((ext_vector_type(2))) float v2f;
typedef __attribute__((ext_vector_type(8))) float v8f;
typedef __attribute__((ext_vector_type(4))) int   v4i;

#define ALPHA 0.2f
#define BS 8
#define NN 1024
#define FIN 256
#define HEADS 8
#define FO 64
#define HF (HEADS * FO) /* 512 */

// ---------------------------------------------------------------------------
// Async global->LDS staging (gfx1250 ASYNCcnt path), with safe fallback.
// Builtin signature (from clang diagnostic): (v4i addrspace(1)*, v4i addrspace(3)*, imm, imm)
// ---------------------------------------------------------------------------
#if __has_builtin(__builtin_amdgcn_global_load_async_to_lds_b128) && \
    __has_builtin(__builtin_amdgcn_s_wait_asynccnt)
#define USE_ASYNC_LDS 1
#define ASYNC_WAIT(n) __builtin_amdgcn_s_wait_asynccnt(n)
#else
#define USE_ASYNC_LDS 0
#define ASYNC_WAIT(n)
#endif

typedef __attribute__((address_space(1))) v4i gv4i;
typedef __attribute__((address_space(3))) v4i lv4i;

__device__ __forceinline__ void copy16_async(const float* g, float* l)
{
#if USE_ASYNC_LDS
    __builtin_amdgcn_global_load_async_to_lds_b128(
        (gv4i*)(float*)g, (lv4i*)l, 0, 0);
#else
    *(float4*)l = *(const float4*)g;
#endif
}

// ---------------------------------------------------------------------------
// Kernel 1: Wh = h @ W   (8192x256) @ (256x512), f32 WMMA 16x16x4
// Grid: (512 m-tiles, 4 n-groups), 256 threads = 8 waves, each wave a 16x16 tile
// ---------------------------------------------------------------------------
__global__ __launch_bounds__(256) void wh_gemm_kernel(
    const float* __restrict__ h, const float* __restrict__ W,
    float* __restrict__ Wh)
{
    __shared__ float hs[16 * (FIN + 4)];   // stride 260 -> bank = (4m+k)%64

    const int mtile = blockIdx.x;          // 0..511
    const int ngrp  = blockIdx.y;          // 0..3
    const int t     = threadIdx.x;
    const int mbase = mtile * 16;

    // stage 16 x 256 h tile via async LDS engine (1024 x 16B chunks)
    for (int c = t; c < 16 * (FIN / 4); c += 256) {
        const int r  = c >> 6;             // / 64 chunks per row
        const int f4 = c & 63;
        copy16_async(h + (mbase + r) * FIN + f4 * 4,
                     &hs[r * (FIN + 4) + f4 * 4]);
    }
    ASYNC_WAIT(0);
    __syncthreads();

    const int wave = t >> 5;
    const int lane = t & 31;
    const int m    = lane & 15;            // A-fragment row
    const int k0   = (lane >> 4) * 2;      // A/B fragment K pair select
    const int ncol = ngrp * 128 + wave * 16 + (lane & 15);

    v8f acc = {};
    for (int k = 0; k < FIN; k += 4) {
        v2f av, bv;
        av.x = hs[m * (FIN + 4) + k + k0];
        av.y = hs[m * (FIN + 4) + k + k0 + 1];
        bv.x = W[(k + k0)     * HF + ncol];
        bv.y = W[(k + k0 + 1) * HF + ncol];
        acc = __builtin_amdgcn_wmma_f32_16x16x4_f32(
            false, av, false, bv, (short)0, acc, false, false);
    }

    // C/D layout: VGPR r -> row r (lanes 0-15) / row 8+r (lanes 16-31)
    const int rofs = (lane >> 4) * 8;
#pragma unroll
    for (int r = 0; r < 8; ++r) {
        Wh[(mbase + r + rofs) * HF + ncol] = acc[r];
    }
}

// ---------------------------------------------------------------------------
// Kernel 2: e_src[b,h,j] = Wh[b,j,h,:] . a_src[h],  e_dst likewise
// ---------------------------------------------------------------------------
__global__ __launch_bounds__(256) void escore_kernel(
    const float* __restrict__ Wh, const float* __restrict__ a,
    float* __restrict__ esrc, float* __restrict__ edst)
{
    const int idx = blockIdx.x * 256 + threadIdx.x;  // 0..65535
    const int b   = idx >> 13;
    const int rem = idx & 8191;
    const int hh  = rem >> 10;
    const int j   = rem & 1023;

    const float4* wv = (const float4*)(Wh + ((size_t)(b * NN + j) * HF + hh * FO));
    const float4* as = (const float4*)(a + hh * 2 * FO);
    const float4* ad = (const float4*)(a + hh * 2 * FO + FO);

    float s = 0.f, d = 0.f;
#pragma unroll
    for (int f = 0; f < FO / 4; ++f) {
        const float4 w4 = wv[f], s4 = as[f], d4 = ad[f];
        s += w4.x * s4.x + w4.y * s4.y + w4.z * s4.z + w4.w * s4.w;
        d += w4.x * d4.x + w4.y * d4.y + w4.z * d4.z + w4.w * d4.w;
    }
    esrc[(b * HEADS + hh) * NN + j] = s;
    edst[(b * HEADS + hh) * NN + j] = d;
}

// ---------------------------------------------------------------------------
// Kernel 3: fused masked-leakyReLU-softmax + (attn @ Wh), flash style.
// Grid: (64 i-tiles, 8 heads, 8 batch), 128 threads = 4 waves.
// Wh j-block tiles double-buffered in LDS via async loads; P tile shared.
// ---------------------------------------------------------------------------
#define BJ   64
#define PSTR 68   /* A-frag reads: bank = (4m + k) % 64, conflict-free      */
#define WSTR 72   /* B-frag reads: 2*72 % 64 = 16 -> half-waves disjoint    */
#define NB   (NN / BJ)

__global__ __launch_bounds__(128) void gat_attn_kernel(
    const float* __restrict__ Wh, const int* __restrict__ adj,
    const float* __restrict__ esrc, const float* __restrict__ edst,
    float* __restrict__ out)
{
    __shared__ float P[16 * PSTR];
    __shared__ float Whs[2][BJ * WSTR];
    __shared__ float pm[16 * 8], ps[16 * 8];
    __shared__ float mrun[16], lrun[16], scl[16], mnw[16], es[16];

    const int itile = blockIdx.x;   // 0..63
    const int hh    = blockIdx.y;   // 0..7
    const int b     = blockIdx.z;   // 0..7
    const int t     = threadIdx.x;
    const int lane  = t & 31;
    const int wave  = t >> 5;

    const float* esr = esrc + (b * HEADS + hh) * NN;
    const float* eds = edst + (b * HEADS + hh) * NN;

    // stage Wh j-block tile: 64 rows x 64 fo-cols = 1024 x 16B chunks / 128 thr
    auto stage = [&](int jb2, int bufi) {
        const float* src = Wh + (size_t)(b * NN + jb2 * BJ) * HF + hh * FO;
        for (int c = t; c < BJ * (FO / 4); c += 128) {   // 8 iters/thread
            const int r  = c >> 4;
            const int f4 = c & 15;
            copy16_async(src + r * HF + f4 * 4, &Whs[bufi][r * WSTR + f4 * 4]);
        }
    };

    if (t < 16) {
        mrun[t] = -3.0e38f;
        lrun[t] = 0.f;
        es[t]   = esr[itile * 16 + t];
    }
    stage(0, 0);                    // prefetch block 0 (8 async instrs/wave)

    const int m    = lane & 15;
    const int k0   = (lane >> 4) * 2;
    const int ncol = hh * FO + wave * 16 + (lane & 15);
    const int rofs = (lane >> 4) * 8;

    v8f acc = {};

    const int si = t >> 3;          // score row this thread owns (0..15)
    const int sj = (t & 7) * 8;     // its 8 consecutive cols

    for (int jb = 0; jb < NB; ++jb) {
        const int j0  = jb * BJ;
        const int buf = jb & 1;

        // issue async stage of NEXT Wh block into the other buffer
        if (jb + 1 < NB) stage(jb + 1, buf ^ 1);

        // --- step 1: raw masked scores in registers + per-thread max -------
        float svv[8];
        {
            const int*   arow = adj + (size_t)(b * NN + itile * 16 + si) * NN + j0 + sj;
            const float* ed   = eds + j0 + sj;
            const float  esi  = es[si];
            float tmax = -3.0e38f;
#pragma unroll
            for (int q = 0; q < 8; ++q) {
                float sv = esi + ed[q];
                sv = sv > 0.f ? sv : ALPHA * sv;
                sv = (arow[q] == 0) ? -3.0e38f : sv;
                svv[q] = sv;
                tmax = fmaxf(tmax, sv);
            }
            pm[si * 8 + (t & 7)] = tmax;
            if (jb + 1 < NB) __builtin_prefetch(arow + BJ, 0, 1);  // next adj
        }
        // current Wh buffer complete when <= next block's 8 asyncs outstanding
        if (jb + 1 < NB) { ASYNC_WAIT(8); } else { ASYNC_WAIT(0); }
        __syncthreads();

        // --- step 2: row max reduce + running-max update (16 threads) ------
        if (t < 16) {
            const float mold = mrun[t];
            float bm = mold;
#pragma unroll
            for (int q = 0; q < 8; ++q) bm = fmaxf(bm, pm[t * 8 + q]);
            mnw[t]  = bm;
            scl[t]  = __expf(mold - bm);
            mrun[t] = bm;
        }
        __syncthreads();

        // --- step 3: parallel exp into P + partial sums; rescale acc -------
        {
            const float mrow = mnw[si];
            float tsum = 0.f;
#pragma unroll
            for (int q = 0; q < 8; ++q) {
                const float p = __expf(svv[q] - mrow);
                P[si * PSTR + sj + q] = p;
                tsum += p;
            }
            ps[si * 8 + (t & 7)] = tsum;
#pragma unroll
            for (int r = 0; r < 8; ++r) acc[r] *= scl[r + rofs];
        }
        __syncthreads();

        // --- step 4: running-sum update (16 threads), then WMMA ------------
        if (t < 16) {
            float bsum = 0.f;
#pragma unroll
            for (int q = 0; q < 8; ++q) bsum += ps[t * 8 + q];
            lrun[t] = lrun[t] * scl[t] + bsum;
        }

        // acc += P(16xBJ) @ Whs(BJ x 16 fo-slice), A from P, B from LDS tile
        const float* wb = &Whs[buf][wave * 16 + (lane & 15)];
        for (int k = 0; k < BJ; k += 4) {
            v2f av, bv;
            av.x = P[m * PSTR + k + k0];
            av.y = P[m * PSTR + k + k0 + 1];
            bv.x = wb[(k + k0)     * WSTR];
            bv.y = wb[(k + k0 + 1) * WSTR];
            acc = __builtin_amdgcn_wmma_f32_16x16x4_f32(
                false, av, false, bv, (short)0, acc, false, false);
        }
        __syncthreads();   // P & lrun consumed before next block overwrites
    }

    // --- epilogue: divide by running sum, write out ------------------------
#pragma unroll
    for (int r = 0; r < 8; ++r) {
        const int row = itile * 16 + r + rofs;
        out[(size_t)(b * NN + row) * HF + ncol] = acc[r] / lrun[r + rofs];
    }
}

// ---------------------------------------------------------------------------
extern "C" void kernel_launch(void* const* d_in, const int* in_sizes, int n_in,
                              void* d_out, int out_size, void* d_ws, size_t ws_size,
                              hipStream_t stream)
{
    const float* h   = (const float*)d_in[0];
    const int*   adj = (const int*)d_in[1];
    const float* W   = (const float*)d_in[2];
    const float* a   = (const float*)d_in[3];
    float*       out = (float*)d_out;

    // workspace layout: Wh (16 MB) | e_src (256 KB) | e_dst (256 KB)
    float* Wh   = (float*)d_ws;
    float* esrc = Wh + (size_t)BS * NN * HF;
    float* edst = esrc + (size_t)BS * HEADS * NN;

    wh_gemm_kernel<<<dim3(512, 4), 256, 0, stream>>>(h, W, Wh);
    escore_kernel<<<256, 256, 0, stream>>>(Wh, a, esrc, edst);
    gat_attn_kernel<<<dim3(NN / 16, HEADS, BS), 128, 0, stream>>>(
        Wh, adj, esrc, edst, out);
}